// HungarianMatcher_34591666602013
// MI455X (gfx1250) — compile-verified
//
#include <hip/hip_runtime.h>
#include <hip/hip_bf16.h>

typedef __attribute__((ext_vector_type(2))) float v2f;
typedef __attribute__((ext_vector_type(8))) float v8f;

#define NB   64
#define NQ   900
#define NC   91
#define NT   300
#define QTILES 57          // ceil(900/16)
#define KCHUNKS 23         // ceil(91/4), column 91 zero-padded
#define PSTR 93            // prob row stride (odd -> no LDS bank conflicts)
#define QSTR 10            // per-query geometry row stride
#define WAVE_LDS (16*PSTR + 16*QSTR)

__global__ __launch_bounds__(128) void matcher_cost_kernel(
    const float* __restrict__ logits,   // [B,Q,C]
    const float* __restrict__ bboxes,   // [B,Q,4] cxcywh
    const float* __restrict__ tboxes,   // [B,T,4] cxcywh
    const int*   __restrict__ tids,     // [B,T]
    float* __restrict__ out)            // [B,Q,T]
{
    __shared__ float lds[4 * WAVE_LDS];
    const int wave = threadIdx.x >> 5;
    const int lane = threadIdx.x & 31;
    float* probs = lds + wave * WAVE_LDS;   // [16][PSTR]: logits tile -> softmax numerators
    float* qrow  = probs + 16 * PSTR;       // [16][QSTR]: x0 y0 x1 y1 cx cy w h inv

    const int tile  = blockIdx.x * 4 + wave;   // grid sized so tile < 3648 always
    const int b     = tile / QTILES;
    const int qtile = (tile % QTILES) * 16;

    const int  qs = lane & 15;                 // query-in-tile / target column
    const bool hi = (lane & 16) != 0;
    const int  qg = min(qtile + qs, NQ - 1);   // clamped global query index

    // ---------- stage 16x91 logits tile to LDS via CDNA5 async global->LDS ----------
    {
        const uint32_t probsBase = (uint32_t)(uintptr_t)(void*)probs;
        const size_t   lbase     = (size_t)b * NQ;
        for (int i = lane; i < 16 * NC; i += 32) {
            const int r = i / NC;
            const int c = i - r * NC;
            const int grow = min(qtile + r, NQ - 1);
            const float*  gp   = logits + (lbase + grow) * NC + c;
            const uint32_t loff = probsBase + (uint32_t)(r * PSTR + c) * 4u;
            asm volatile("global_load_async_to_lds_b32 %0, %1, off"
                         :: "v"(loff), "v"((unsigned long long)(uintptr_t)gp)
                         : "memory");
        }
        asm volatile("s_wait_asynccnt 0x0" ::: "memory");
        __builtin_amdgcn_wave_barrier();
    }

    // ---------- cooperative softmax from LDS: 2 lanes per query split C ----------
    const int c0 = hi ? 46 : 0;
    const int c1 = hi ? NC : 46;
    float mx = -3.0e38f;
    for (int c = c0; c < c1; ++c) mx = fmaxf(mx, probs[qs * PSTR + c]);
    mx = fmaxf(mx, __shfl_xor(mx, 16, 32));
    float sum = 0.0f;
    for (int c = c0; c < c1; ++c) {
        float e = __expf(probs[qs * PSTR + c] - mx);
        probs[qs * PSTR + c] = e;               // in-place: each slot owned by one lane
        sum += e;
    }
    if (hi) probs[qs * PSTR + 91] = 0.0f;       // pad column for K=88..91 chunk
    sum += __shfl_xor(sum, 16, 32);
    const float inv = 1.0f / sum;

    if (!hi) {                                  // per-query geometry row
        const float* bb = bboxes + ((size_t)b * NQ + qg) * 4;
        float cx = bb[0], cy = bb[1], w = bb[2], h = bb[3];
        float* r = qrow + qs * QSTR;
        r[0] = cx - 0.5f * w;  r[1] = cy - 0.5f * h;
        r[2] = cx + 0.5f * w;  r[3] = cy + 0.5f * h;
        r[4] = cx; r[5] = cy; r[6] = w; r[7] = h;
        r[8] = inv;
    }
    __syncthreads();

    // ---------- hoist A fragments (t-invariant prob rows) ----------
    // 16x4 f32 A layout: lanes 0-15 hold M=lane {K0 in v0, K1 in v1},
    //                    lanes 16-31 hold M=lane-16 {K2 in v0, K3 in v1}
    float a0[KCHUNKS], a1[KCHUNKS];
    #pragma unroll
    for (int kc = 0; kc < KCHUNKS; ++kc) {
        const int kb = kc * 4 + (hi ? 2 : 0);
        a0[kc] = probs[qs * PSTR + kb];
        a1[kc] = probs[qs * PSTR + kb + 1];
    }

    // ---------- sweep target tiles ----------
    for (int tt = 0; tt < NT; tt += 16) {
        const int  t      = tt + qs;
        const bool tvalid = (t < NT);
        const int  tc     = min(t, NT - 1);
        const int  tid    = tids[(size_t)b * NT + tc];

        const float* tb = tboxes + ((size_t)b * NT + tc) * 4;
        const float tcx = tb[0], tcy = tb[1], tw = tb[2], th = tb[3];
        const float tx0 = tcx - 0.5f * tw, ty0 = tcy - 0.5f * th;
        const float tx1 = tcx + 0.5f * tw, ty1 = tcy + 0.5f * th;
        const float ta  = tw * th;

        // one-hot GEMM: acc[m][n] = prob_numerator[qtile+m][tids[tt+n]]
        v8f acc = {};
        #pragma unroll
        for (int kc = 0; kc < KCHUNKS; ++kc) {
            const int kb = kc * 4 + (hi ? 2 : 0);
            v2f a;  a.x = a0[kc];  a.y = a1[kc];
            v2f bm;
            bm.x = (tid == kb)     ? 1.0f : 0.0f;
            bm.y = (tid == kb + 1) ? 1.0f : 0.0f;
            acc = __builtin_amdgcn_wmma_f32_16x16x4_f32(
                      false, a, false, bm, (short)0, acc, false, false);
        }

        // epilogue: D VGPR v -> (M = v or v+8, N = lane&15)
        #pragma unroll
        for (int v = 0; v < 8; ++v) {
            const int m = v + (hi ? 8 : 0);
            const int q = qtile + m;
            const float* r = qrow + m * QSTR;          // broadcast reads
            const float qx0 = r[0], qy0 = r[1], qx1 = r[2], qy1 = r[3];
            const float qcx = r[4], qcy = r[5], qw = r[6], qh = r[7];
            const float qinv = r[8];
            const float qa = qw * qh;

            const float l1 = fabsf(qcx - tcx) + fabsf(qcy - tcy)
                           + fabsf(qw  - tw ) + fabsf(qh  - th );

            const float ix = fmaxf(fminf(qx1, tx1) - fmaxf(qx0, tx0), 0.0f);
            const float iy = fmaxf(fminf(qy1, ty1) - fmaxf(qy0, ty0), 0.0f);
            const float inter = ix * iy;
            const float uni   = qa + ta - inter;
            const float iou   = inter * __builtin_amdgcn_rcpf(uni + 1e-7f);

            const float ex = fmaxf(fmaxf(qx1, tx1) - fminf(qx0, tx0), 0.0f);
            const float ey = fmaxf(fmaxf(qy1, ty1) - fminf(qy0, ty0), 0.0f);
            const float ae = ex * ey;
            const float giou = iou - (ae - uni) * __builtin_amdgcn_rcpf(ae + 1e-7f);

            const float res = (-acc[v] * qinv) + 5.0f * l1 - 2.0f * giou;
            if (tvalid && q < NQ)
                out[((size_t)b * NQ + q) * NT + t] = res;
        }
    }
}

extern "C" void kernel_launch(void* const* d_in, const int* in_sizes, int n_in,
                              void* d_out, int out_size, void* d_ws, size_t ws_size,
                              hipStream_t stream) {
    (void)in_sizes; (void)n_in; (void)d_ws; (void)ws_size; (void)out_size;
    const float* logits = (const float*)d_in[0];
    const float* bboxes = (const float*)d_in[1];
    const float* tboxes = (const float*)d_in[2];
    const int*   tids   = (const int*)d_in[3];
    float* out = (float*)d_out;

    // 64 batches * 57 q-tiles = 3648 waves; 4 waves/block -> 912 blocks
    matcher_cost_kernel<<<912, 128, 0, stream>>>(logits, bboxes, tboxes, tids, out);
}